// EdgeDecoder_77429670412869
// MI455X (gfx1250) — compile-verified
//
#include <hip/hip_runtime.h>

typedef float v2f __attribute__((ext_vector_type(2)));
typedef float v8f __attribute__((ext_vector_type(8)));

#define DNODE 50      // per-node embedding dim
#define KCH   25      // K = 2*DNODE = 100 -> 25 chunks of 4
#define HID   32      // hidden dim (2 N-tiles of 16)

// LDS-staged W1 in WMMA-B-fragment order:
// Bs[(c*2 + t)*32 + lane] = { W1[(4c+off)*32 + n], W1[(4c+off+1)*32 + n] }
//   with off = 2*(lane>>4), n = t*16 + (lane&15)
__global__ __launch_bounds__(256) void edge_decoder_wmma(
    const float* __restrict__ z,
    const int*   __restrict__ eidx,   // [2, E] flat: src row then dst row
    const float* __restrict__ W1,     // [100, 32] row-major
    const float* __restrict__ b1,     // [32]
    const float* __restrict__ W2,     // [32, 1]
    const float* __restrict__ b2,     // [1]
    float*       __restrict__ out,    // [E]
    int E, int numTiles)
{
    __shared__ v2f Bs[KCH * 2 * 32];  // 12.8 KB

    const int lane = threadIdx.x & 31;
    const int wave = threadIdx.x >> 5;
    const int wavesPerBlock = blockDim.x >> 5;
    const int half = lane >> 4;       // 0: K pair {0,1} / M 0..7 ; 1: K pair {2,3} / M 8..15
    const int l15  = lane & 15;
    const int koff = half * 2;

    // ---- cooperative staging of W1 into B-fragment layout ----
    for (int i = threadIdx.x; i < KCH * 2 * 32; i += blockDim.x) {
        const int l  = i & 31;
        const int ct = i >> 5;
        const int c  = ct >> 1;
        const int t  = ct & 1;
        const int n  = t * 16 + (l & 15);
        const int k0 = 4 * c + 2 * (l >> 4);
        v2f v;
        v[0] = W1[k0 * HID + n];
        v[1] = W1[(k0 + 1) * HID + n];
        Bs[i] = v;
    }
    __syncthreads();

    // per-lane second-layer constants (loop invariant)
    const float b1n0 = b1[l15];
    const float b1n1 = b1[16 + l15];
    const float w2n0 = W2[l15];
    const float w2n1 = W2[16 + l15];
    const float bias2 = b2[0];

    int tile = blockIdx.x * wavesPerBlock + wave;
    const int tileStride = gridDim.x * wavesPerBlock;

    for (; tile < numTiles; tile += tileStride) {
        const int base = tile * 16;
        int e = base + l15;
        if (e >= E) e = E - 1;                 // uniform-EXEC tail clamp
        const int isrc = eidx[e];
        const int idst = eidx[E + e];
        const float* __restrict__ srow = z + (long long)isrc * DNODE;
        const float* __restrict__ drow = z + (long long)idst * DNODE;

        v8f c0 = {};
        v8f c1 = {};
#pragma unroll
        for (int c = 0; c < KCH; ++c) {
            const int f = 4 * c + koff;        // feature index of this lane's pair
            const float* p = (f < DNODE) ? (srow + f) : (drow + (f - DNODE));
            const v2f a = *(const v2f*)p;      // float2 gather: A fragment half
            const v2f bf0 = Bs[(c * 2 + 0) * 32 + lane];
            const v2f bf1 = Bs[(c * 2 + 1) * 32 + lane];
            c0 = __builtin_amdgcn_wmma_f32_16x16x4_f32(false, a, false, bf0,
                                                       (short)0, c0, false, false);
            c1 = __builtin_amdgcn_wmma_f32_16x16x4_f32(false, a, false, bf1,
                                                       (short)0, c1, false, false);
        }

        // epilogue: out[m] = relu(h + b1) . W2 + b2, reduce over N within 16-lane halves
#pragma unroll
        for (int r = 0; r < 8; ++r) {
            float h0 = c0[r] + b1n0; h0 = h0 > 0.f ? h0 : 0.f;
            float h1 = c1[r] + b1n1; h1 = h1 > 0.f ? h1 : 0.f;
            float v = h0 * w2n0 + h1 * w2n1;
            v += __shfl_xor(v, 8, 32);
            v += __shfl_xor(v, 4, 32);
            v += __shfl_xor(v, 2, 32);
            v += __shfl_xor(v, 1, 32);
            if (l15 == 0) {
                const int oe = base + r + 8 * half;
                if (oe < E) out[oe] = v + bias2;
            }
        }
    }
}

extern "C" void kernel_launch(void* const* d_in, const int* in_sizes, int n_in,
                              void* d_out, int out_size, void* d_ws, size_t ws_size,
                              hipStream_t stream) {
    const float* z    = (const float*)d_in[0];
    const int*   eidx = (const int*)  d_in[1];
    const float* W1   = (const float*)d_in[2];
    const float* b1   = (const float*)d_in[3];
    const float* W2   = (const float*)d_in[4];
    const float* b2   = (const float*)d_in[5];
    float* out = (float*)d_out;

    const int E = in_sizes[1] / 2;             // edge_label_index is [2, E]
    const int numTiles = (E + 15) / 16;
    const int wavesPerBlock = 8;               // 256 threads
    int blocks = (numTiles + wavesPerBlock - 1) / wavesPerBlock;
    if (blocks > 3072) blocks = 3072;          // persistent grid, amortize LDS staging
    edge_decoder_wmma<<<blocks, 256, 0, stream>>>(z, eidx, W1, b1, W2, b2, out,
                                                  E, numTiles);
}